// GPEncoder_40321152975257
// MI455X (gfx1250) — compile-verified
//
#include <hip/hip_runtime.h>
#include <math.h>

typedef __attribute__((ext_vector_type(2))) float v2f;
typedef __attribute__((ext_vector_type(8))) float v8f;

#define HW    4096
#define O_YS  786432
#define O_M   1572864
#define O_SP  1573248
#define O_PS  51904896
#define K_LOG2PI 58.81206625f   /* 32 * ln(2*pi) */

__device__ __forceinline__ v8f wmma4(v2f a, v2f b, v8f c) {
    // D(16x16,f32) = A(16x4,f32) * B(4x16,f32) + C
    return __builtin_amdgcn_wmma_f32_16x16x4_f32(false, a, false, b, (short)0, c, false, false);
}

__device__ __forceinline__ float wred(float v) {
#pragma unroll
    for (int off = 16; off > 0; off >>= 1) v += __shfl_xor(v, off);
    return v;
}

__device__ __forceinline__ void lds_fence() {
    asm volatile("s_wait_dscnt 0" ::: "memory");
}

// ---------------- simple float4 copy ----------------
__global__ void copy_f4_kernel(const float4* __restrict__ src, float4* __restrict__ dst, int n4) {
    int i = blockIdx.x * blockDim.x + threadIdx.x;
    if (i < n4) dst[i] = src[i];
}

// ---------------- S (lb, c[1024], p[4096]) -> Sp (lb, p, c) ----------------
__global__ void transpose_kernel(const float* __restrict__ S, float* __restrict__ Sp) {
    __shared__ float tile[32][33];
    int lb = blockIdx.z;
    int p0 = blockIdx.x * 32;
    int c0 = blockIdx.y * 32;
    int tx = threadIdx.x, ty = threadIdx.y;
    size_t sbase = (size_t)lb * 1024 * 4096;
#pragma unroll
    for (int i = 0; i < 4; ++i) {
        int c = c0 + ty + i * 8;
        tile[ty + i * 8][tx] = S[sbase + (size_t)c * 4096 + p0 + tx];
    }
    __syncthreads();
    size_t dbase = (size_t)lb * 4096 * 1024;
#pragma unroll
    for (int i = 0; i < 4; ++i) {
        int p = p0 + ty + i * 8;
        Sp[dbase + (size_t)p * 1024 + c0 + tx] = tile[tx][ty + i * 8];
    }
}

// ---------------- per-pixel math: one wave per pixel ----------------
// LDS per wave (floats): bufA 1056 (D^-1, later Sl/Chol), bufR 1056 (R),
// bufC 544 (E -> Syx -> Q), bufD 272 (Sxx -> Sxx^-1), bufT 32 (t/z).
#define WAVE_LDS 2960

__global__ void __launch_bounds__(128) gp_math_kernel(
    const float* __restrict__ xx, const float* __restrict__ m,
    const float* __restrict__ R,  const float* __restrict__ D,
    float* out)
{
    __shared__ float smem[4 * WAVE_LDS];
    const int wid  = threadIdx.x >> 5;
    const int ln   = threadIdx.x & 31;
    const int half = ln >> 4;
    const int r16  = ln & 15;

    float* bufA = smem + wid * WAVE_LDS;
    float* bufR = bufA + 1056;
    float* bufC = bufR + 1056;
    float* bufD = bufC + 544;
    float* bufT = bufD + 272;

    const int pix = blockIdx.x * 4 + wid;   // 0..49151
    const int lb  = pix >> 12;              // l*4+b
    const int p   = pix & 4095;

    const size_t mat_base = ((size_t)lb * HW + p) * 1024;
    const float* Dp = D + mat_base;
    const float* Rp = R + mat_base;

    // ---- load D row 'ln' into registers; stage R (coalesced) into LDS (pad 33)
    float a[32];
#pragma unroll
    for (int j4 = 0; j4 < 8; ++j4) {
        float4 dv = *(const float4*)(Dp + (size_t)ln * 32 + j4 * 4);
        a[j4 * 4 + 0] = dv.x; a[j4 * 4 + 1] = dv.y;
        a[j4 * 4 + 2] = dv.z; a[j4 * 4 + 3] = dv.w;
        int f0 = (j4 * 32 + ln) * 4;
        float4 rv = *(const float4*)(Rp + f0);
        int adr = (f0 >> 5) * 33 + (f0 & 31);
        bufR[adr + 0] = rv.x; bufR[adr + 1] = rv.y;
        bufR[adr + 2] = rv.z; bufR[adr + 3] = rv.w;
    }

    // ---- in-place Gauss-Jordan inversion of D (SPD, no pivoting), row-per-lane
#pragma unroll
    for (int k = 0; k < 32; ++k) {
        float pk   = __shfl(a[k], k);
        float pinv = 1.0f / pk;
        float f    = a[k];
#pragma unroll
        for (int j = 0; j < 32; ++j) {
            float akj    = __shfl(a[j], k);
            float scaled = akj * pinv;
            float newk   = (j == k) ? pinv : scaled;
            float upd    = (j == k) ? (-f * pinv) : (a[j] - f * scaled);
            a[j] = (ln == k) ? newk : upd;
        }
    }
#pragma unroll
    for (int j = 0; j < 32; ++j) bufA[ln * 33 + j] = a[j];
    lds_fence();

    // ---- E = D^-1 (32x32) x R^T (32x16)  -> bufC (32 rows, stride 17)
    v8f e0 = {}; v8f e1 = {};
#pragma unroll
    for (int kk = 0; kk < 8; ++kk) {
        int ka = kk * 4 + 2 * half;
        v2f b;  b.x  = bufR[r16 * 33 + ka];        b.y  = bufR[r16 * 33 + ka + 1]; // B[k][n]=R[n][k]
        v2f a0; a0.x = bufA[r16 * 33 + ka];        a0.y = bufA[r16 * 33 + ka + 1];
        v2f a1; a1.x = bufA[(16 + r16) * 33 + ka]; a1.y = bufA[(16 + r16) * 33 + ka + 1];
        e0 = wmma4(a0, b, e0);
        e1 = wmma4(a1, b, e1);
    }
#pragma unroll
    for (int v = 0; v < 8; ++v) {
        bufC[(v + 8 * half) * 17 + r16]      = e0[v];
        bufC[(16 + v + 8 * half) * 17 + r16] = e1[v];
    }
    lds_fence();

    // ---- S1c = R (32x32) x E (32x16): s0 = Sxx rows, s1 = Syx rows
    v8f s0 = {}; v8f s1 = {};
#pragma unroll
    for (int kk = 0; kk < 8; ++kk) {
        int ka = kk * 4 + 2 * half;
        v2f b;  b.x  = bufC[ka * 17 + r16];        b.y  = bufC[(ka + 1) * 17 + r16];
        v2f a0; a0.x = bufR[r16 * 33 + ka];        a0.y = bufR[r16 * 33 + ka + 1];
        v2f a1; a1.x = bufR[(16 + r16) * 33 + ka]; a1.y = bufR[(16 + r16) * 33 + ka + 1];
        s0 = wmma4(a0, b, s0);
        s1 = wmma4(a1, b, s1);
    }
#pragma unroll
    for (int v = 0; v < 8; ++v) bufD[(v + 8 * half) * 17 + r16] = s0[v];  // Sxx
    lds_fence();
#pragma unroll
    for (int v = 0; v < 8; ++v) bufC[(v + 8 * half) * 17 + r16] = s1[v];  // Syx (bufC reuse)
    lds_fence();

    // ---- Gauss-Jordan invert Sxx (16x16); lanes 16-31 mirror rows 0-15
    float q[16];
#pragma unroll
    for (int j = 0; j < 16; ++j) q[j] = bufD[r16 * 17 + j];
#pragma unroll
    for (int k = 0; k < 16; ++k) {
        float pk   = __shfl(q[k], k);
        float pinv = 1.0f / pk;
        float f    = q[k];
#pragma unroll
        for (int j = 0; j < 16; ++j) {
            float qkj    = __shfl(q[j], k);
            float scaled = qkj * pinv;
            float newk   = (j == k) ? pinv : scaled;
            float upd    = (j == k) ? (-f * pinv) : (q[j] - f * scaled);
            q[j] = (r16 == k) ? newk : upd;
        }
    }
    if (ln < 16) {
#pragma unroll
        for (int j = 0; j < 16; ++j) bufD[ln * 17 + j] = q[j];
    }
    lds_fence();

    // ---- Q = Syx (16x16) x Sxx^-1 (16x16)
    v8f qt = {};
#pragma unroll
    for (int kk = 0; kk < 4; ++kk) {
        int ka = kk * 4 + 2 * half;
        v2f aa; aa.x = bufC[r16 * 17 + ka]; aa.y = bufC[r16 * 17 + ka + 1];
        v2f bb; bb.x = bufD[ka * 17 + r16]; bb.y = bufD[(ka + 1) * 17 + r16];
        qt = wmma4(aa, bb, qt);
    }
#pragma unroll
    for (int v = 0; v < 8; ++v) bufC[(v + 8 * half) * 17 + r16] = qt[v];   // Q rows
    lds_fence();

    // ---- yy = my + Q (xx - mx)
    const float* mlb = m + lb * 32;
    float mv = mlb[ln];
    volatile float* vT = bufT;
    float tv = 0.0f;
    if (ln < 16) {
        float xv = xx[((size_t)lb * 16 + ln) * HW + p];
        tv = xv - mv;
        vT[ln] = tv;
    }
    lds_fence();
    if (ln < 16) {
        float acc = mlb[16 + ln];
#pragma unroll
        for (int n = 0; n < 16; ++n) acc += bufC[ln * 17 + n] * vT[n];
        out[O_YS + ((size_t)lb * 16 + ln) * HW + p] = acc;
        vT[16 + ln] = acc;   // z upper half
    }
    lds_fence();
    float diff = (ln < 16) ? tv : (vT[ln] - mv);

    // ---- stage Sl (from transposed output region, coalesced) into bufA
    const float* Sl = out + O_SP + mat_base;
#pragma unroll
    for (int j4 = 0; j4 < 8; ++j4) {
        int f0 = (j4 * 32 + ln) * 4;
        float4 sv = *(const float4*)(Sl + f0);
        int adr = (f0 >> 5) * 33 + (f0 & 31);
        bufA[adr + 0] = sv.x; bufA[adr + 1] = sv.y;
        bufA[adr + 2] = sv.z; bufA[adr + 3] = sv.w;
    }
    lds_fence();

    // ---- Cholesky (lower) of Sl in LDS, wave-ordered DS ops
    volatile float* vA = bufA;
    for (int k = 0; k < 32; ++k) {
        float dkk  = vA[k * 33 + k];
        float lkk  = sqrtf(dkk);
        float linv = 1.0f / lkk;
        float aik  = vA[ln * 33 + k];
        float lik  = aik * linv;
        if (ln == k)      vA[k * 33 + k]  = lkk;
        else if (ln > k)  vA[ln * 33 + k] = lik;
        for (int j = k + 1; j < 32; ++j) {
            float ljk = vA[j * 33 + k];
            if (ln >= j) vA[ln * 33 + j] = vA[ln * 33 + j] - lik * ljk;
        }
    }
    float logdet = 2.0f * wred(logf(vA[ln * 33 + ln]));

    // ---- forward solve L y = diff
    float y = diff;
    for (int k = 0; k < 32; ++k) {
        float lkk = vA[k * 33 + k];
        float yk  = __shfl(y, k) / lkk;
        if (ln == k)     y = yk;
        else if (ln > k) y = y - vA[ln * 33 + k] * yk;
    }
    // ---- back solve L^T x = y
    float xsol = y;
    for (int k = 31; k >= 0; --k) {
        float lkk = vA[k * 33 + k];
        float xk  = __shfl(xsol, k) / lkk;
        if (ln == k)     xsol = xk;
        else if (ln < k) xsol = xsol - vA[k * 33 + ln] * xk;
    }
    float quad = wred(diff * xsol);
    if (ln == 0)
        out[O_PS + (size_t)lb * HW + p] = -0.5f * (quad + logdet + K_LOG2PI);
}

extern "C" void kernel_launch(void* const* d_in, const int* in_sizes, int n_in,
                              void* d_out, int out_size, void* d_ws, size_t ws_size,
                              hipStream_t stream) {
    (void)in_sizes; (void)n_in; (void)out_size; (void)d_ws; (void)ws_size;
    // inputs: 0=x (unused), 1=tgt (unused), 2=xx, 3=m, 4=R, 5=D, 6=S
    const float* xx = (const float*)d_in[2];
    const float* m  = (const float*)d_in[3];
    const float* R  = (const float*)d_in[4];
    const float* D  = (const float*)d_in[5];
    const float* S  = (const float*)d_in[6];
    float* out = (float*)d_out;

    // xs = copy(xx): 786432 floats = 196608 float4
    copy_f4_kernel<<<(196608 + 255) / 256, 256, 0, stream>>>(
        (const float4*)xx, (float4*)out, 196608);
    // m passthrough: 384 floats = 96 float4
    copy_f4_kernel<<<1, 96, 0, stream>>>(
        (const float4*)m, (float4*)(out + O_M), 96);
    // Sp = transpose(S): per lb a 1024x4096 -> 4096x1024 transpose
    transpose_kernel<<<dim3(128, 32, 12), dim3(32, 8), 0, stream>>>(S, out + O_SP);
    // per-pixel math (reads Sp region of d_out written above; stream-ordered)
    gp_math_kernel<<<12288, 128, 0, stream>>>(xx, m, R, D, out);
}